// STAGNNAggregator_Optimized_72610717106762
// MI455X (gfx1250) — compile-verified
//
#include <hip/hip_runtime.h>
#include <hip/hip_bf16.h>

typedef __attribute__((ext_vector_type(2))) float v2f;
typedef __attribute__((ext_vector_type(8))) float v8f;

#define T_DIM 8
#define NINT 30   // valid (t,dt) pairs: 1+2+3+4+5*4

__constant__ int c_t[NINT]  = {0,1,1,2,2,2,3,3,3,3,4,4,4,4,4,5,5,5,5,5,6,6,6,6,6,7,7,7,7,7};
__constant__ int c_dt[NINT] = {0,0,1,0,1,2,0,1,2,3,0,1,2,3,4,0,1,2,3,4,0,1,2,3,4,0,1,2,3,4};

// Monotone order-preserving float<->uint map (exact atomic max on floats).
__device__ __forceinline__ unsigned f2key(float f) {
    unsigned b = __float_as_uint(f);
    return (b & 0x80000000u) ? ~b : (b | 0x80000000u);
}
__device__ __forceinline__ float key2f(unsigned k) {
    unsigned b = (k & 0x80000000u) ? (k & 0x7FFFFFFFu) : ~k;
    return __uint_as_float(b);
}

#define HPAD 68   // 16B-aligned row stride (272B), bank-conflict-free column reads

// ---------------------------------------------------------------------------
// Fused Q / K_base / V GEMM: D[80000x64] = H[80000x64] x W^T, f32 WMMA.
// Block = 384 threads = 12 waves; wave w -> (mat = w/4, ntile = w%4).
// H tile staged LDS via async global->LDS DMA (wave 0), A from LDS, B global.
// ---------------------------------------------------------------------------
__global__ __launch_bounds__(384) void qkv_gemm(
    const float* __restrict__ H,
    const float* __restrict__ Wq, const float* __restrict__ Wk,
    const float* __restrict__ Wv,
    float* __restrict__ Q, float* __restrict__ Kb, float* __restrict__ V) {
    __shared__ float hs[16 * HPAD];
    const int mtile = blockIdx.x;
    const float* Hrow = H + (size_t)mtile * 16 * 64;

    if (threadIdx.x < 32) {
        // 16 rows x 256B, copied as 256 x 16B chunks by wave 0, async to LDS.
        #pragma unroll
        for (int i = 0; i < 8; ++i) {
            const int c   = threadIdx.x + 32 * i;    // 0..255
            const int r   = c >> 4, q16 = c & 15;
            const unsigned lds_off =
                (unsigned)(size_t)&hs[0] + (unsigned)(r * HPAD * 4 + q16 * 16);
            const unsigned long long gaddr =
                (unsigned long long)(const void*)(Hrow + r * 64 + q16 * 4);
            asm volatile("global_load_async_to_lds_b128 %0, %1, off"
                         :: "v"(lds_off), "v"(gaddr) : "memory");
        }
        asm volatile("s_wait_asynccnt 0" ::: "memory");
    }
    __syncthreads();

    const int wv   = threadIdx.x >> 5;  // 0..11
    const int lane = threadIdx.x & 31;
    const int mat  = wv >> 2;           // 0=Q, 1=K, 2=V
    const int nt   = wv & 3;            // 0..3 (16-col tile)
    const float* W = (mat == 0) ? Wq : (mat == 1 ? Wk : Wv);
    const int wstride = (mat == 1) ? 72 : 64;  // Wk is [64 x 72]; use first 64 cols
    float* Dst = (mat == 0) ? Q : (mat == 1 ? Kb : V);

    const int l  = lane & 15;   // M index (A) / N index (B,D)
    const int hh = lane >> 4;   // K-pair selector
    const float* wrow = W + (size_t)(nt * 16 + l) * wstride;

    v8f acc = {};
    #pragma unroll
    for (int ks = 0; ks < 16; ++ks) {
        const int kb = ks * 4 + hh * 2;
        v2f a; a.x = hs[l * HPAD + kb]; a.y = hs[l * HPAD + kb + 1];
        v2f b; b.x = wrow[kb];          b.y = wrow[kb + 1];
        acc = __builtin_amdgcn_wmma_f32_16x16x4_f32(
            false, a, false, b, (short)0, acc, false, false);
    }

    float* dp = Dst + (size_t)mtile * 16 * 64;
    #pragma unroll
    for (int r = 0; r < 8; ++r) {
        const int row = r + hh * 8;                 // C/D layout: lanes16-31 hold M+8
        dp[(size_t)row * 64 + nt * 16 + l] = acc[r];
    }
}

// ---------------------------------------------------------------------------
// K_pe[dt][o] = sum_j pe[dt][j] * Wk[o][64+j]   (tiny)
// ---------------------------------------------------------------------------
__global__ void kpe_k(const float* __restrict__ Wk, float* __restrict__ Kpe) {
    const int o = threadIdx.x;  // 64 threads
    for (int dt = 0; dt <= 4; ++dt) {
        const float f = (float)dt;
        float pe[8];
        pe[0] = __expf(-f * 0.25f);    // tau = 4
        pe[1] = __expf(-f * 0.0625f);  // tau = 16
        pe[2] = __sinf(f);  pe[3] = __sinf(f * 0.5f);  pe[4] = __sinf(f * 0.25f);
        pe[5] = __cosf(f);  pe[6] = __cosf(f * 0.5f);  pe[7] = __cosf(f * 0.25f);
        float acc = 0.f;
        #pragma unroll
        for (int j = 0; j < 8; ++j) acc += pe[j] * Wk[(size_t)o * 72 + 64 + j];
        Kpe[dt * 64 + o] = acc;
    }
}

// lgate[tn, h] = log(mean_{16} S[tn, h*16..] + 1e-6)
__global__ void gate_k(const float* __restrict__ S, float* __restrict__ lgate, int TN) {
    const int idx = blockIdx.x * blockDim.x + threadIdx.x;
    if (idx >= TN * 4) return;
    const int tn = idx >> 2, h = idx & 3;
    const float4* p = (const float4*)(S + (size_t)tn * 64 + h * 16);
    float s = 0.f;
    #pragma unroll
    for (int j = 0; j < 4; ++j) {
        float4 v = p[j];
        s += v.x + v.y + v.z + v.w;
    }
    lgate[idx] = logf(s * (1.f / 16.f) + 1e-6f);
}

__global__ void init_k(float* __restrict__ out, float* __restrict__ denom,
                       unsigned* __restrict__ smax, int TN) {
    const long long idx = (long long)blockIdx.x * blockDim.x + threadIdx.x;
    if (idx < (long long)TN * 64) out[idx] = 0.f;
    if (idx < (long long)TN * 4) { denom[idx] = 0.f; smax[idx] = 0u; }
}

// ---------------------------------------------------------------------------
// Pass 1: per (interaction,edge) row, per-head score -> atomic segment max.
// Optionally stores per-row scores (sstore != null) to skip pass-2 regather.
// ---------------------------------------------------------------------------
__global__ __launch_bounds__(256) void score_max(
    const float* __restrict__ Q, const float* __restrict__ Kb,
    const float* __restrict__ Kpe, const float* __restrict__ lgate,
    const float* __restrict__ rbias, const int* __restrict__ ei,
    unsigned* __restrict__ smax, float* __restrict__ sstore,
    int N, int E, long long M) {
    const int lane = threadIdx.x & 31;
    const int half = lane >> 4, l = lane & 15;
    const long long row = (long long)blockIdx.x * 16 + (threadIdx.x >> 5) * 2 + half;
    if (row >= M) return;
    const int inter = (int)(row / E);
    const int e     = (int)(row - (long long)inter * E);
    const int t = c_t[inter], dt = c_dt[inter], tp = t - dt;
    const int s = ei[e], d2 = ei[E + e];

    const float4 q4 = *(const float4*)(Q + ((size_t)t * N + d2) * 64 + l * 4);
    float4 k4       = *(const float4*)(Kb + ((size_t)tp * N + s) * 64 + l * 4);
    const float4 p4 = *(const float4*)(Kpe + dt * 64 + l * 4);
    k4.x += p4.x; k4.y += p4.y; k4.z += p4.z; k4.w += p4.w;

    float ps = q4.x * k4.x + q4.y * k4.y + q4.z * k4.z + q4.w * k4.w;
    ps += __shfl_xor(ps, 1, 32);
    ps += __shfl_xor(ps, 2, 32);
    const int h = l >> 2;
    const float score = ps * 0.25f + rbias[dt] + lgate[((size_t)tp * N + s) * 4 + h];
    if ((l & 3) == 0) {
        atomicMax(smax + ((size_t)t * N + d2) * 4 + h, f2key(score));
        if (sstore) sstore[row * 4 + h] = score;
    }
}

// ---------------------------------------------------------------------------
// Pass 2: w = exp(score - smax), atomic denom + weighted-V scatter into d_out.
// Score read from sstore if available, else recomputed (Q/K regather).
// ---------------------------------------------------------------------------
__global__ __launch_bounds__(256) void score_acc(
    const float* __restrict__ Q, const float* __restrict__ Kb,
    const float* __restrict__ Kpe, const float* __restrict__ lgate,
    const float* __restrict__ rbias, const int* __restrict__ ei,
    const unsigned* __restrict__ smax, float* __restrict__ denom,
    const float* __restrict__ V, float* __restrict__ out,
    const float* __restrict__ sstore, int N, int E, long long M) {
    const int lane = threadIdx.x & 31;
    const int half = lane >> 4, l = lane & 15;
    const long long row = (long long)blockIdx.x * 16 + (threadIdx.x >> 5) * 2 + half;
    if (row >= M) return;
    const int inter = (int)(row / E);
    const int e     = (int)(row - (long long)inter * E);
    const int t = c_t[inter], dt = c_dt[inter], tp = t - dt;
    const int s = ei[e], d2 = ei[E + e];
    const size_t srcb = ((size_t)tp * N + s) * 64;
    const size_t dstb = ((size_t)t * N + d2) * 64;
    const int h = l >> 2;

    float score;
    if (sstore) {
        score = sstore[row * 4 + h];
    } else {
        const float4 q4 = *(const float4*)(Q + dstb + l * 4);
        float4 k4       = *(const float4*)(Kb + srcb + l * 4);
        const float4 p4 = *(const float4*)(Kpe + dt * 64 + l * 4);
        k4.x += p4.x; k4.y += p4.y; k4.z += p4.z; k4.w += p4.w;
        float ps = q4.x * k4.x + q4.y * k4.y + q4.z * k4.z + q4.w * k4.w;
        ps += __shfl_xor(ps, 1, 32);
        ps += __shfl_xor(ps, 2, 32);
        score = ps * 0.25f + rbias[dt] + lgate[((size_t)tp * N + s) * 4 + h];
    }
    const float sm = key2f(smax[((size_t)t * N + d2) * 4 + h]);
    const float w = __expf(score - sm);

    if ((l & 3) == 0)
        atomicAdd(denom + ((size_t)t * N + d2) * 4 + h, w);

    const float4 v4 = *(const float4*)(V + srcb + l * 4);
    float* op = out + dstb + l * 4;
    atomicAdd(op + 0, w * v4.x);
    atomicAdd(op + 1, w * v4.y);
    atomicAdd(op + 2, w * v4.z);
    atomicAdd(op + 3, w * v4.w);
}

__global__ void norm_k(float* __restrict__ out, const float* __restrict__ denom, int TN) {
    const long long idx = (long long)blockIdx.x * blockDim.x + threadIdx.x;
    if (idx >= (long long)TN * 64) return;
    const long long tn = idx >> 6;
    const int h = ((int)idx >> 4) & 3;
    out[idx] = out[idx] / fmaxf(denom[tn * 4 + h], 1e-12f);
}

// ---------------------------------------------------------------------------
extern "C" void kernel_launch(void* const* d_in, const int* in_sizes, int n_in,
                              void* d_out, int out_size, void* d_ws, size_t ws_size,
                              hipStream_t stream) {
    const float* H  = (const float*)d_in[0];
    const float* S  = (const float*)d_in[1];
    const int*   ei = (const int*)d_in[2];
    // d_in[3] = time_idx (arange, unused)
    const float* Wq = (const float*)d_in[4];
    const float* Wk = (const float*)d_in[5];
    const float* Wv = (const float*)d_in[6];
    const float* rb = (const float*)d_in[7];

    const int TN = in_sizes[0] / 64;   // T*N = 80000
    const int N  = TN / T_DIM;
    const int E  = in_sizes[2] / 2;
    const long long M = (long long)NINT * E;

    float* Q     = (float*)d_ws;
    float* Kb    = Q + (size_t)TN * 64;
    float* V     = Kb + (size_t)TN * 64;
    float* Kpe   = V + (size_t)TN * 64;
    float* lgate = Kpe + 512;
    float* denom = lgate + (size_t)TN * 4;
    unsigned* smax = (unsigned*)(denom + (size_t)TN * 4);
    float* sbase   = (float*)(smax + (size_t)TN * 4);

    // Use stored-score path only if workspace is big enough for M*4 floats.
    const size_t base_floats = (size_t)TN * 64 * 3 + 512 + (size_t)TN * 4 * 3;
    const size_t need = (base_floats + (size_t)M * 4) * sizeof(float);
    float* sstore = (ws_size >= need) ? sbase : nullptr;

    float* out = (float*)d_out;

    hipLaunchKernelGGL(init_k, dim3((TN * 64 + 255) / 256), dim3(256), 0, stream,
                       out, denom, smax, TN);
    hipLaunchKernelGGL(kpe_k, dim3(1), dim3(64), 0, stream, Wk, Kpe);
    hipLaunchKernelGGL(gate_k, dim3((TN * 4 + 255) / 256), dim3(256), 0, stream,
                       S, lgate, TN);
    hipLaunchKernelGGL(qkv_gemm, dim3(TN / 16), dim3(384), 0, stream,
                       H, Wq, Wk, Wv, Q, Kb, V);
    const unsigned blocks = (unsigned)((M + 15) / 16);
    hipLaunchKernelGGL(score_max, dim3(blocks), dim3(256), 0, stream,
                       Q, Kb, Kpe, lgate, rb, ei, smax, sstore, N, E, M);
    hipLaunchKernelGGL(score_acc, dim3(blocks), dim3(256), 0, stream,
                       Q, Kb, Kpe, lgate, rb, ei, smax, denom, V, out, sstore, N, E, M);
    hipLaunchKernelGGL(norm_k, dim3((TN * 64 + 255) / 256), dim3(256), 0, stream,
                       out, denom, TN);
}